// Easy2hardLoss_1958505087538
// MI455X (gfx1250) — compile-verified
//
#include <hip/hip_runtime.h>
#include <cstdint>

#define BAN_TARGET 3

typedef __attribute__((ext_vector_type(2))) float v2f;
typedef __attribute__((ext_vector_type(8))) float v8f;

// ---------------------------------------------------------------------------
// CDNA5 async global->LDS copy (GVS mode: SGPR64 base + VGPR32 byte offset).
// ISA 15.18.3 opcode 98 GLOBAL_LOAD_ASYNC_TO_LDS_B128, tracked by ASYNCcnt.
// vdst = LDS byte address VGPR, vaddr = 32-bit global byte offset, saddr = base.
// ---------------------------------------------------------------------------
__device__ __forceinline__ void async_copy_b128(uint32_t lds_byte_addr,
                                                uint32_t gvoff_bytes,
                                                const float* sbase) {
  asm volatile("global_load_async_to_lds_b128 %0, %1, %2"
               :: "v"(lds_byte_addr), "v"(gvoff_bytes), "s"(sbase)
               : "memory");
}

template <int N>
__device__ __forceinline__ void wait_asynccnt() {
#if __has_builtin(__builtin_amdgcn_s_wait_asynccnt)
  __builtin_amdgcn_s_wait_asynccnt(N);
#else
  asm volatile("s_wait_asynccnt %0" :: "i"(N) : "memory");
#endif
}

// ---- online log-sum-exp helpers -------------------------------------------
__device__ __forceinline__ void online8(float& m, float& s,
                                        const float4& a, const float4& b) {
  float cm = fmaxf(fmaxf(fmaxf(a.x, a.y), fmaxf(a.z, a.w)),
                   fmaxf(fmaxf(b.x, b.y), fmaxf(b.z, b.w)));
  float mn = fmaxf(m, cm);
  float e = __expf(a.x - mn) + __expf(a.y - mn) + __expf(a.z - mn) +
            __expf(a.w - mn) + __expf(b.x - mn) + __expf(b.y - mn) +
            __expf(b.z - mn) + __expf(b.w - mn);
  s = s * __expf(m - mn) + e;   // m==-inf on first use -> exp(-inf)=0, s=0: ok
  m = mn;
}

__device__ __forceinline__ void online1(float& m, float& s, float x) {
  float mn = fmaxf(m, x);
  s = s * __expf(m - mn) + __expf(x - mn);
  m = mn;
}

__device__ __forceinline__ void lse_merge(float& m, float& s, float om, float os) {
  float mn = fmaxf(m, om);
  float ta = (s  != 0.f) ? s  * __expf(m  - mn) : 0.f;
  float tb = (os != 0.f) ? os * __expf(om - mn) : 0.f;
  m = mn;
  s = ta + tb;
}

// ---------------------------------------------------------------------------
// Kernel 1: one workgroup (8 wave32s) per row. Async-LDS double-buffered
// streaming log-sum-exp; per-row loss to workspace.
// ---------------------------------------------------------------------------
__global__ void __launch_bounds__(256)
ce_row_kernel(const float* __restrict__ logits, const int* __restrict__ target,
              const int* __restrict__ epoch_p, float* __restrict__ row_loss,
              int C) {
  constexpr int K      = 6;           // float4 per thread per chunk (6 async b128 / wave / chunk)
  constexpr int CH_F4  = 256 * K;     // 1536 float4 per chunk
  constexpr int CH_EL  = CH_F4 * 4;   // 6144 floats per chunk
  __shared__ float4 buf[2][CH_F4];    // 48 KB double buffer
  __shared__ float red_m[8], red_s[8];

  const int row = blockIdx.x;
  const int t   = threadIdx.x;
  const float* rowbase = logits + (size_t)row * (size_t)C;

  const int nchunk     = C / CH_EL;
  const int tail_start = nchunk * CH_EL;

  float m = -__builtin_inff(), s = 0.f;

  const int tg = target[row];
  float xt = 0.f;
  if (t == 0) xt = rowbase[tg];

  auto issue = [&](int c, int bsel) {
#pragma unroll
    for (int j = 0; j < K; ++j) {
      const uint32_t idx = (uint32_t)(j * 256 + t);                 // f4 index in chunk
      const uint32_t go  = ((uint32_t)c * (uint32_t)CH_F4 + idx) * 16u;
      // generic LDS address: low 32 bits == LDS byte offset (aperture layout)
      const uint32_t la  = (uint32_t)(uintptr_t)&buf[bsel][idx];
      async_copy_b128(la, go, rowbase);
    }
  };

  if (nchunk > 0) issue(0, 0);
  for (int c = 0; c < nchunk; ++c) {
    const int bsel = c & 1;
    if (c + 1 < nchunk) {
      issue(c + 1, bsel ^ 1);   // keep next chunk in flight
      wait_asynccnt<K>();       // oldest K (current chunk) complete, in order
    } else {
      wait_asynccnt<0>();
    }
    // Each thread reads back exactly the float4s its own wave copied in,
    // so no workgroup barrier is needed in the streaming loop.
#pragma unroll
    for (int j = 0; j < K; j += 2) {
      float4 a = buf[bsel][j * 256 + t];
      float4 b = buf[bsel][(j + 1) * 256 + t];
      online8(m, s, a, b);
    }
  }

  // tail: direct bounds-checked global reads (never read past the row)
  for (int i = tail_start + t; i < C; i += 256) online1(m, s, rowbase[i]);

  // wave32 reduction
#pragma unroll
  for (int o = 16; o > 0; o >>= 1) {
    float om = __shfl_xor(m, o, 32);
    float os = __shfl_xor(s, o, 32);
    lse_merge(m, s, om, os);
  }
  const int lane = t & 31, wv = t >> 5;
  if (lane == 0) { red_m[wv] = m; red_s[wv] = s; }
  __syncthreads();

  if (t == 0) {
    float M = red_m[0], S = red_s[0];
#pragma unroll
    for (int w = 1; w < 8; ++w) lse_merge(M, S, red_m[w], red_s[w]);
    float loss = (M + logf(S)) - xt;

    const int ep = *epoch_p;
    int mode = 0;
    if (ep <= 80) { if (ep % 5 != 0) mode = 1; }
    else if (ep % 5 != 0) mode = 2;
    if ((mode == 1 && tg == BAN_TARGET) || (mode == 2 && tg != BAN_TARGET))
      loss = 0.f;
    row_loss[row] = loss;
  }
}

// ---------------------------------------------------------------------------
// Kernel 2: deterministic mean of B per-row losses. Final 32-partial
// reduction via V_WMMA_F32_16X16X4_F32 (A = partials in 16x4 f32 layout,
// B = ones -> D rows are group sums). EXEC is all-ones for wave 0.
// ---------------------------------------------------------------------------
__global__ void __launch_bounds__(256)
mean_kernel(const float* __restrict__ rl, float* __restrict__ outp,
            int B, float inv_denom) {
  __shared__ float p[256];
  const int t = threadIdx.x;
  float s = 0.f;
  for (int i = t; i < B; i += 256) s += rl[i];
  p[t] = s;
  __syncthreads();

  if (t < 32) {                 // wave-uniform: whole wave 0 active, EXEC=all 1s
    float a = 0.f;
#pragma unroll
    for (int j = 0; j < 8; ++j) a += p[t * 8 + j];

#if __has_builtin(__builtin_amdgcn_wmma_f32_16x16x4_f32)
    // A 16x4 f32 layout: VGPR0 = K=0 (lanes 0-15) / K=2 (lanes 16-31);
    // VGPR1 = K=1 / K=3. Put partial in slot 0, zero in slot 1; B = ones.
    v2f A;  A[0] = a;   A[1] = 0.f;
    v2f Bm; Bm[0] = 1.f; Bm[1] = 1.f;
    v8f Cz = {};
    v8f D = __builtin_amdgcn_wmma_f32_16x16x4_f32(
        false, A, false, Bm, (short)0, Cz, false, false);
    // D[r] (lanes 0-15) = a of lane r + a of lane r+16; lanes 16-31 cover rows 8-15.
    float t8 = D[0] + D[1] + D[2] + D[3] + D[4] + D[5] + D[6] + D[7];
    float tot = t8 + __shfl_xor(t8, 16, 32);
#else
    float tot = a;
#pragma unroll
    for (int o = 16; o > 0; o >>= 1) tot += __shfl_xor(tot, o, 32);
#endif
    if (t == 0) *outp = tot * inv_denom;
  }
}

// ---------------------------------------------------------------------------
extern "C" void kernel_launch(void* const* d_in, const int* in_sizes, int n_in,
                              void* d_out, int out_size, void* d_ws, size_t ws_size,
                              hipStream_t stream) {
  const float* logits = (const float*)d_in[0];
  const int*   target = (const int*)d_in[1];
  const int*   epoch  = (const int*)d_in[2];
  const int B = in_sizes[1];
  const int C = in_sizes[0] / B;

  float* row_loss = (float*)d_ws;   // B floats of scratch

  ce_row_kernel<<<B, 256, 0, stream>>>(logits, target, epoch, row_loss, C);
  mean_kernel<<<1, 256, 0, stream>>>(row_loss, (float*)d_out, B, 1.0f / (float)B);
}